// FullSearchVectorQuantization_28879360098764
// MI455X (gfx1250) — compile-verified
//
#include <hip/hip_runtime.h>

typedef __attribute__((ext_vector_type(2))) float v2f;
typedef __attribute__((ext_vector_type(8))) float v8f;

#define NCB 4
#define CB_SIZE 1024
#define NDIM 8
#define NPOINT 32768
#define NTILE_K (CB_SIZE / 16)      // 64 column tiles
#define TILES_PER_CB (NPOINT / 16)  // 2048 row tiles per codebook

// ---------------------------------------------------------------------------
// Kernel 1: zero-fill the one_hot region with 128-bit stores (pure BW).
// ---------------------------------------------------------------------------
__global__ void __launch_bounds__(256) vq_zero_onehot(float* __restrict__ oh, long n4) {
  long i = (long)blockIdx.x * blockDim.x + threadIdx.x;
  long stride = (long)gridDim.x * blockDim.x;
  float4 z = make_float4(0.f, 0.f, 0.f, 0.f);
  float4* p = (float4*)oh;
  for (; i < n4; i += stride) p[i] = z;
}

// ---------------------------------------------------------------------------
// Kernel 2: one wave per 16-point tile of one codebook.
//   dist = ||x||^2/8 + ||c||^2/8 - (x.c)/4, cross term via WMMA f32 16x16x4.
//   A is pre-scaled by -0.25 so acc already holds -(x.c)/4.
//   All norms computed lane-locally from full-row b128 loads (no in-loop LDS).
// ---------------------------------------------------------------------------
__global__ void __launch_bounds__(256) vq_kernel(const float* __restrict__ x,
                                                 const float* __restrict__ cb,
                                                 float* __restrict__ xhat,
                                                 float* __restrict__ onehot,
                                                 float* __restrict__ dist) {
  const int lane = threadIdx.x & 31;
  const int wid  = threadIdx.x >> 5;
  const int gw   = blockIdx.x * 8 + wid;          // 8192 waves total, exact
  const int a    = gw / TILES_PER_CB;             // codebook index
  const int tile = gw % TILES_PER_CB;
  const int n0   = tile * 16;

  const int  row16 = lane & 15;                   // M (for A) / N (for B)
  const int  half  = lane >> 4;                   // 0 or 1
  const bool hi    = (half != 0);                 // K-pair {2,3} vs {0,1}
  const int  base8 = half * 8;                    // C/D row base for this half

  // ---- Load full x row (2 x b128), norms lane-locally ----
  const float4* xr4 =
      (const float4*)(x + ((size_t)a * NPOINT + (size_t)(n0 + row16)) * NDIM);
  float4 p0 = xr4[0];
  float4 p1 = xr4[1];
  float x2q = 0.125f * (p0.x * p0.x + p0.y * p0.y + p0.z * p0.z + p0.w * p0.w +
                        p1.x * p1.x + p1.y * p1.y + p1.z * p1.z + p1.w * p1.w);

  // Broadcast the 8 row-norms this lane's C slots will need (one-time).
  float x2v[8];
#pragma unroll
  for (int v = 0; v < 8; ++v) x2v[v] = __shfl(x2q, base8 + v);

  // A operand: dims {kb, kb+1, kb+4, kb+5}, scaled by -2/ndim = -0.25.
  v2f a0, a1;
  a0.x = -0.25f * (hi ? p0.z : p0.x);
  a0.y = -0.25f * (hi ? p0.w : p0.y);
  a1.x = -0.25f * (hi ? p1.z : p1.x);
  a1.y = -0.25f * (hi ? p1.w : p1.y);

  float best[8];
  int   bidx[8];
#pragma unroll
  for (int v = 0; v < 8; ++v) { best[v] = 3.4e38f; bidx[v] = 0; }

  const float* cbbase   = cb + (size_t)a * CB_SIZE * NDIM;
  float*       distbase = dist + ((size_t)a * NPOINT + (size_t)n0) * CB_SIZE;

  for (int t = 0; t < NTILE_K; ++t) {
    const int col = t * 16 + row16;
    const float4* cr4 = (const float4*)(cbbase + (size_t)col * NDIM);
    float4 q0 = cr4[0];
    float4 q1 = cr4[1];
    // ||c[col]||^2 / ndim, lane-local (no cross-lane dependency in the loop).
    float c2q = 0.125f * (q0.x * q0.x + q0.y * q0.y + q0.z * q0.z + q0.w * q0.w +
                          q1.x * q1.x + q1.y * q1.y + q1.z * q1.z + q1.w * q1.w);
    // B operand: dims {kb, kb+1, kb+4, kb+5} (loop-invariant selects).
    v2f b0, b1;
    b0.x = hi ? q0.z : q0.x;
    b0.y = hi ? q0.w : q0.y;
    b1.x = hi ? q1.z : q1.x;
    b1.y = hi ? q1.w : q1.y;

    v8f acc = {};
    acc = __builtin_amdgcn_wmma_f32_16x16x4_f32(false, a0, false, b0,
                                                (short)0, acc, false, false);
    acc = __builtin_amdgcn_wmma_f32_16x16x4_f32(false, a1, false, b1,
                                                (short)0, acc, false, false);

#pragma unroll
    for (int v = 0; v < 8; ++v) {
      float d = acc[v] + x2v[v] + c2q;  // dist(row=base8+v, col)
      distbase[(size_t)(base8 + v) * CB_SIZE + col] = d;
      if (d < best[v]) { best[v] = d; bidx[v] = col; }  // first-min semantics
    }
  }

  // ---- Cross-lane argmin within each 16-lane half (xor masks stay in-half) ----
#pragma unroll
  for (int m = 1; m < 16; m <<= 1) {
#pragma unroll
    for (int v = 0; v < 8; ++v) {
      float ov = __shfl_xor(best[v], m);
      int   oi = __shfl_xor(bidx[v], m);
      if (ov < best[v] || (ov == best[v] && oi < bidx[v])) {
        best[v] = ov; bidx[v] = oi;
      }
    }
  }

  // ---- Writers: lanes 0..7 -> rows 0..7, lanes 16..23 -> rows 8..15 ----
  const int s = lane & 15;
  if (s < 8) {
    const int row = base8 + s;
    const int idx = bidx[s];
    const size_t rflat = (size_t)a * NPOINT + (size_t)(n0 + row);
    onehot[rflat * CB_SIZE + idx] = 1.0f;
    const float4* csrc = (const float4*)(cbbase + (size_t)idx * NDIM);
    float4* xdst = (float4*)(xhat + rflat * NDIM);
    xdst[0] = csrc[0];
    xdst[1] = csrc[1];
  }
}

// ---------------------------------------------------------------------------
extern "C" void kernel_launch(void* const* d_in, const int* in_sizes, int n_in,
                              void* d_out, int out_size, void* d_ws, size_t ws_size,
                              hipStream_t stream) {
  (void)in_sizes; (void)n_in; (void)out_size; (void)d_ws; (void)ws_size;
  const float* x  = (const float*)d_in[0];
  const float* cb = (const float*)d_in[1];

  float* out    = (float*)d_out;
  float* xhat   = out;                                      // 4*32768*8
  float* onehot = xhat + (size_t)NCB * NPOINT * NDIM;       // 4*32768*1024
  float* dist   = onehot + (size_t)NCB * NPOINT * CB_SIZE;  // 4*32768*1024

  // Zero one_hot with b128 stores (main kernel then scatters the 1.0s).
  const long n4 = ((long)NCB * NPOINT * CB_SIZE) / 4;       // 33,554,432 float4
  vq_zero_onehot<<<131072, 256, 0, stream>>>(onehot, n4);

  // 8192 waves = 1024 blocks x 8 waves; exact tiling, no bounds checks.
  vq_kernel<<<1024, 256, 0, stream>>>(x, cb, xhat, onehot, dist);
}